// Attention_40510131535961
// MI455X (gfx1250) — compile-verified
//
#include <hip/hip_runtime.h>
#include <hip/hip_bf16.h>

// Output of the reference collapses: softmax over a size-1 axis => attention
// weights are all 1.0, so context[b,0,e] = sum_s encode_output[b,s,e].
// W1/b1/W2/b2/hidden_state are dead. This is a pure bandwidth-bound
// column-sum (256 MiB streamed, ~11.5 us floor at 23.3 TB/s).
//
// We perform the reduction with V_WMMA_F32_16X16X4_F32 using an all-ones A
// matrix: D = ones(16x4) * B(4x16) + C accumulates 4 rows x 16 cols of
// encode_output per instruction into the f32 accumulator. Since A is all
// ones, the K<->slot mapping of B is irrelevant (sum is commutative), so no
// cross-lane shuffles are needed; the column<->lane mapping (lane L holds
// column L&15) follows the ISA's 32-bit B/C/D striping exactly.

typedef __attribute__((ext_vector_type(2))) float v2f;
typedef __attribute__((ext_vector_type(8))) float v8f;

#define BB      32      // batch
#define SS      2048    // seq len
#define EE      1024    // ENC_OUT
#define WAVES   8
#define THREADS (WAVES * 32)
#define SROWS   (SS / WAVES)   // 256 rows per wave

__global__ __launch_bounds__(THREADS)
void Attention_40510131535961_rowsum_wmma(const float* __restrict__ enc,
                                          float* __restrict__ out) {
  const int tile = blockIdx.x;          // 16-column tile index (0..63)
  const int b    = blockIdx.y;          // batch (0..31)
  const int tid  = threadIdx.x;
  const int wave = tid >> 5;
  const int lane = tid & 31;
  const int n    = lane & 15;           // column within tile (B/C/D lane map)
  const int half = lane >> 4;           // 0: lanes 0-15, 1: lanes 16-31

  // Base pointer for this lane's column; rows advance by EE floats.
  const float* base = enc + ((size_t)b * SS) * EE + (size_t)tile * 16 + n;
  const int sBeg = wave * SROWS;

  // A = all ones (16x4 f32 -> 2 VGPRs per lane). Exact K mapping irrelevant.
  const v2f a = {1.0f, 1.0f};
  v8f c = {};                           // 16x16 f32 accumulator (8 VGPRs)

  // Lane covers rows {sBeg+half, sBeg+2+half} -> bijection onto the 4 K rows.
  const float* p0 = base + (size_t)(sBeg + half) * EE;
  const float* p1 = base + (size_t)(sBeg + 2 + half) * EE;

  #pragma unroll 4
  for (int s = 0; s < SROWS; s += 4) {
    // Stream ahead: data is single-use, keep it from thrashing caches.
    __builtin_prefetch(p0 + (size_t)32 * EE, 0, 0);
    v2f bm;
    bm.x = __builtin_nontemporal_load(p0);   // row s+half     (64B/half-wave)
    bm.y = __builtin_nontemporal_load(p1);   // row s+2+half   (64B/half-wave)
    // 8 args: (neg_a, A, neg_b, B, c_mod, C, reuse_a, reuse_b)
    c = __builtin_amdgcn_wmma_f32_16x16x4_f32(false, a, false, bm,
                                              (short)0, c, false, false);
    p0 += (size_t)4 * EE;
    p1 += (size_t)4 * EE;
  }

  // Every row of D holds the same column sums (A rows identical).
  // Lanes 0-15, VGPR0 => row M=0, columns 0..15 of this wave's partial sum.
  __shared__ float red[WAVES * 16];
  if (lane < 16) red[wave * 16 + n] = c[0];
  __syncthreads();

  if (tid < 16) {
    float sum = 0.0f;
    #pragma unroll
    for (int w = 0; w < WAVES; ++w) sum += red[w * 16 + tid];
    out[(size_t)b * EE + (size_t)tile * 16 + tid] = sum;
  }
}

extern "C" void kernel_launch(void* const* d_in, const int* in_sizes, int n_in,
                              void* d_out, int out_size, void* d_ws, size_t ws_size,
                              hipStream_t stream) {
  (void)in_sizes; (void)n_in; (void)d_ws; (void)ws_size; (void)out_size;
  const float* enc = (const float*)d_in[0];  // encode_output [32,2048,1024]
  // d_in[1..5] (hidden_state, W1, b1, W2, b2) are dead: softmax over a
  // singleton axis makes every attention weight exactly 1.0.
  float* out = (float*)d_out;                // [32,1,1024]

  dim3 grid(EE / 16, BB);   // 64 x 32 blocks
  dim3 block(THREADS);      // 8 waves of 32
  Attention_40510131535961_rowsum_wmma<<<grid, block, 0, stream>>>(enc, out);
}